// SignalingModel_31791347925219
// MI455X (gfx1250) — compile-verified
//
#include <hip/hip_runtime.h>

// Problem dimensions (fixed by the reference)
#define NN    4096   // network nodes
#define NB    16     // batch
#define NIN_  128    // input ligands
#define NOUT_ 256    // output TFs
#define STEPS 60
#define LEAK  0.01f

typedef float v2f __attribute__((ext_vector_type(2)));
typedef float v8f __attribute__((ext_vector_type(8)));

// Michaelis-Menten-like activation, faithful to the reference:
//   fx = leaky_relu(x); shifted = 0.5*(fx-0.5); mask = shifted<0
//   gated = fx + 10*mask; right = 0.5 + shifted/gated
//   result = mask ? fx : right
__device__ __forceinline__ float mml_act(float x) {
    float fx      = (x >= 0.0f) ? x : LEAK * x;
    float shifted = 0.5f * (fx - 0.5f);
    bool  mask    = shifted < 0.0f;
    float gated   = mask ? (fx + 10.0f) : fx;   // divisor always safe
    float right   = 0.5f + shifted / gated;
    return mask ? fx : right;
}

// ---- setup: Xbias[n][b] = bias[n] (layout [N,B]); XT0 = 0 ------------------
__global__ __launch_bounds__(256) void init_xbias_kernel(
    const float* __restrict__ bias, float* __restrict__ Xbias,
    float* __restrict__ XcT) {
    int t = blockIdx.x * 256 + threadIdx.x;   // [0, NN*NB)
    int n = t >> 4;                           // NB == 16
    Xbias[t] = bias[n];
    XcT[t]   = 0.0f;
}

// ---- setup: Xbias[in_idx[i]][b] = bias[in_idx[i]] + w_in[i]*X_in[b][i] -----
__global__ __launch_bounds__(256) void scatter_in_kernel(
    const float* __restrict__ X_in, const float* __restrict__ w_in,
    const int* __restrict__ in_idx, const float* __restrict__ bias,
    float* __restrict__ Xbias) {
    int t = blockIdx.x * 256 + threadIdx.x;   // [0, NIN*NB)
    if (t >= NIN_ * NB) return;
    int i = t >> 4;          // ligand
    int b = t & 15;          // batch
    int node = in_idx[i];    // permutation slice -> distinct nodes (pure set)
    Xbias[node * NB + b] = bias[node] + w_in[i] * X_in[b * NIN_ + i];
}

// ---- one fixed-point step: XnT = MML(W @ Xc + Xbias), X kept as [B, N] -----
// Grid: 256 blocks (one per 16-row output tile). Block: 8 waves, 8-way
// split-K (512 K each, 128 WMMAs), LDS reduce, fused bias + activation.
// fp32 WMMA 16x16x4: A = W tile (16x4), B = X tile (4x16), C/D 16x16 f32.
// With X transposed, both A and B fragments are one aligned b64 load/lane.
__global__ __launch_bounds__(256) void step_kernel(
    const float* __restrict__ W,  const float* __restrict__ XcT,
    const float* __restrict__ Xb, float* __restrict__ XnT) {
    __shared__ float red[8][32][8];           // 8 KB partial accumulators

    // readfirstlane -> wave id & K bounds live in SGPRs -> scalar loop control
    const int wave = __builtin_amdgcn_readfirstlane(threadIdx.x) >> 5;
    const int lane = threadIdx.x & 31;
    const int half = lane >> 4;               // 0: lanes 0-15, 1: lanes 16-31
    const int lid  = lane & 15;
    const int mBase = blockIdx.x * 16;

    const int kBegin = wave * (NN / 8);       // 512-wide K chunk per wave
    const int kEnd   = kBegin + (NN / 8);

    // A lane layout (ISA 16x4 f32): row mBase+lid, cols k + half*2 + {0,1}
    const float* arow = W   + (size_t)(mBase + lid) * NN + half * 2;
    // B lane layout (4x16 f32): rows k + half*2 + {0,1}, col lid
    //   == XT[lid][k + half*2 .. +1]  -> contiguous 8 bytes
    const float* brow = XcT + (size_t)lid * NN + half * 2;

    v8f acc = {0.f, 0.f, 0.f, 0.f, 0.f, 0.f, 0.f, 0.f};

    #pragma unroll 8
    for (int k = kBegin; k < kEnd; k += 4) {
        v2f a = *(const v2f*)(arow + k);      // 8B aligned b64
        v2f b = *(const v2f*)(brow + k);      // 8B aligned b64
        // D = A*B + C (8 args: neg_a, A, neg_b, B, c_mod, C, reuse_a, reuse_b)
        acc = __builtin_amdgcn_wmma_f32_16x16x4_f32(
            false, a, false, b, (short)0, acc, false, false);
    }

    #pragma unroll
    for (int v = 0; v < 8; ++v) red[wave][lane][v] = acc[v];
    __syncthreads();

    if (wave == 0) {
        #pragma unroll
        for (int v = 0; v < 8; ++v) {
            float s = 0.0f;
            #pragma unroll
            for (int w2 = 0; w2 < 8; ++w2) s += red[w2][lane][v];
            // D layout: VGPR v -> row mBase + v + half*8, col lid (batch)
            int row = mBase + v + half * 8;
            float z = s + Xb[row * NB + lid];
            XnT[(size_t)lid * NN + row] = mml_act(z);
        }
    }
}

// ---- output: out[b][j] = w_out[j] * XfT[b][out_idx[j]] + b_out[j] ----------
__global__ __launch_bounds__(256) void gather_out_kernel(
    const float* __restrict__ XfT, const float* __restrict__ w_out,
    const float* __restrict__ b_out, const int* __restrict__ out_idx,
    float* __restrict__ out) {
    int t = blockIdx.x * 256 + threadIdx.x;   // [0, NB*NOUT)
    int b = t >> 8;                           // NOUT == 256
    int j = t & 255;
    out[t] = w_out[j] * XfT[(size_t)b * NN + out_idx[j]] + b_out[j];
}

extern "C" void kernel_launch(void* const* d_in, const int* in_sizes, int n_in,
                              void* d_out, int out_size, void* d_ws, size_t ws_size,
                              hipStream_t stream) {
    const float* X_in   = (const float*)d_in[0];   // [B, NIN]
    const float* W      = (const float*)d_in[1];   // [N, N] row-major
    const float* bias   = (const float*)d_in[2];   // [N, 1]
    const float* w_in   = (const float*)d_in[3];   // [NIN]
    const float* w_out  = (const float*)d_in[4];   // [NOUT]
    const float* b_out  = (const float*)d_in[5];   // [NOUT]
    const int*   in_idx = (const int*)d_in[6];     // [NIN]
    const int*   out_idx= (const int*)d_in[7];     // [NOUT]
    float* out = (float*)d_out;                    // [B, NOUT]

    // Workspace: Xbias [N,B] | X ping [B,N] | X pong [B,N]  (768 KB total)
    float* Xbias = (float*)d_ws;
    float* Xc = Xbias + NN * NB;
    float* Xn = Xc    + NN * NB;

    init_xbias_kernel<<<NN * NB / 256, 256, 0, stream>>>(bias, Xbias, Xc);
    scatter_in_kernel<<<(NIN_ * NB + 255) / 256, 256, 0, stream>>>(
        X_in, w_in, in_idx, bias, Xbias);

    for (int s = 0; s < STEPS; ++s) {
        step_kernel<<<NN / 16, 256, 0, stream>>>(W, Xc, Xbias, Xn);
        float* tmp = Xc; Xc = Xn; Xn = tmp;       // final state ends up in Xc
    }

    gather_out_kernel<<<NB * NOUT_ / 256, 256, 0, stream>>>(
        Xc, w_out, b_out, out_idx, out);
}